// SigLayer_46084999086737
// MI455X (gfx1250) — compile-verified
//
#include <hip/hip_runtime.h>

typedef __attribute__((ext_vector_type(2))) float v2f;
typedef __attribute__((ext_vector_type(4))) float v4f;
typedef __attribute__((ext_vector_type(8))) float v8f;

#define DD     12
#define LPATH  256
#define NROW3  1728        // 12^3
#define TPB    384         // 12 waves (wave32)
#define TPW    9           // 108 M-tiles / 12 waves
#define NGRP   64          // 64 groups * 4 steps = 256 slots (254 real + 2 zero-pad)

__global__ __launch_bounds__(TPB)
void sig_scan_wmma(const float* __restrict__ inp, float* __restrict__ out, int nbatch) {
    __shared__ float P[LPATH * DD];      // 3072 f: whole path tile for this batch
    __shared__ float s3buf[NROW3 * 4];   // 6912 f: s3 rows, K-interleaved [i3*4 + kk]
    __shared__ float zbuf[16 * 4];       // 64 f:  z rows (N padded to 16), [n*4 + kk]

    const int tid  = threadIdx.x;
    const int b    = blockIdx.x;
    const int w    = tid >> 5;
    const int lane = tid & 31;
    const int n    = lane & 15;               // N column (valid if < 12)
    const int k0   = (lane & 16) >> 3;        // 0 or 2: K pair held by this half-wave
    const int hi   = (lane & 16) ? 8 : 0;     // C/D row offset for upper half-wave

    const float* gpath = inp + (size_t)b * (LPATH * DD);

    // ---- load path tile into LDS (768 float4, 2 per thread) ----
    {
        const v4f* g4 = (const v4f*)gpath;
        v4f* p4 = (v4f*)P;
#pragma unroll
        for (int i = 0; i < 2; ++i)
            p4[tid + i * TPB] = g4[tid + i * TPB];
    }
    if (tid < 64) zbuf[tid] = 0.0f;           // cols 12..15 stay 0 forever
    __syncthreads();

    // ---- init carry = restricted_exp(z0), z0 = P[1]-P[0] ----
    // A4 accumulator tiles (C of WMMA): A4_init[i3][j] = A3_init[i3] * z0[j]/4,
    // A3_init[i3] = z0[a]*z0[b]*z0[c]/6  with i3 = a*144 + b*12 + c.
    v8f acc[TPW];
    {
        float z0n = (n < DD) ? (P[DD + n] - P[n]) * 0.25f : 0.0f;
#pragma unroll
        for (int t = 0; t < TPW; ++t) {
            const int m = w * TPW + t;
#pragma unroll
            for (int r = 0; r < 8; ++r) {
                int i3 = 16 * m + r + hi;
                int aa = i3 / 144, bb = (i3 / 12) % 12, cc = i3 % 12;
                float A3i = (P[DD + aa] - P[aa]) * (P[DD + bb] - P[bb]) *
                            (P[DD + cc] - P[cc]) * (1.0f / 6.0f);
                acc[t][r] = A3i * z0n;
            }
        }
    }

    // ---- per-thread private scan state: half an i2-block each (threads 0..287) ----
    const int half  = tid & 1;
    const int i2    = tid >> 1;               // valid for tid < 288
    const int rbase = half * 6;               // first owned j within the i2 block
    const int ia    = i2 / 12, ib = i2 % 12;
    float a2own = 0.0f;
    float a3[6] = {};
    float pcur[6] = {};
    if (tid < 288) {
        float za0 = P[DD + ia] - P[ia];
        float zb0 = P[DD + ib] - P[ib];
        a2own = za0 * zb0 * 0.5f;             // A2 = z0 (x) z0 / 2
#pragma unroll
        for (int rl = 0; rl < 6; ++rl) {
            float z0r = P[DD + rbase + rl] - P[rbase + rl];
            a3[rl] = a2own * z0r * (1.0f / 3.0f);   // A3 = A2 (x) z0 / 3
            pcur[rl] = P[DD + rbase + rl];          // row 1 (prev row of step g=0)
        }
    }

    // ---- scan: 254 steps in 64 groups of 4; A4 deferred to one K=4 WMMA GEMM/group ----
    for (int grp = 0; grp < NGRP; ++grp) {
#pragma unroll
        for (int kk = 0; kk < 4; ++kk) {
            const int g = grp * 4 + kk;            // increment rows (g+1, g+2)
            if (g < 254) {
                const float* Pg1 = &P[(g + 1) * DD];
                const float* Pg2 = &P[(g + 2) * DD];
                if (tid < 288) {
                    float z[6];
#pragma unroll
                    for (int rl = 0; rl < 6; ++rl) {
                        float pn = Pg2[rbase + rl];
                        z[rl] = pn - pcur[rl];
                        pcur[rl] = pn;
                    }
                    float za  = Pg2[ia] - Pg1[ia];
                    float zb  = Pg2[ib] - Pg1[ib];
                    float A1a = Pg1[ia] - P[ia];               // telescoped A1 (old)
                    // k=4 chain and k=3 chain level-2 terms (use OLD A2)
                    float c2  = a2own + (A1a + za * 0.25f)         * (zb * (1.0f / 3.0f));
                    float c2p = a2own + (A1a + za * (1.0f / 3.0f)) * (zb * 0.5f);
                    a2own    += (A1a + za * 0.5f) * zb;            // A2 += (A1+z/2)(x)z
                    float* s3p = &s3buf[(i2 * 12 + rbase) * 4 + kk];
#pragma unroll
                    for (int rl = 0; rl < 6; ++rl) {
                        s3p[rl * 4] = a3[rl] + c2 * (z[rl] * 0.5f); // s3 = A3 + c2 (x) z/2
                        a3[rl]     += c2p * z[rl];                  // A3 += c2' (x) z
                    }
                }
                if (tid < DD) zbuf[tid * 4 + kk] = Pg2[tid] - Pg1[tid];
            } else {
                if (tid < DD) zbuf[tid * 4 + kk] = 0.0f;            // pad step: z = 0
            }
        }
        __syncthreads();
        // A4 += S3(1728x4) x Z(4x16): stage all A operands first, then burst WMMAs
        v2f bv = *(const v2f*)&zbuf[n * 4 + k0];
        v2f av[TPW];
#pragma unroll
        for (int t = 0; t < TPW; ++t) {
            const int m = w * TPW + t;
            av[t] = *(const v2f*)&s3buf[(16 * m + n) * 4 + k0];
        }
#pragma unroll
        for (int t = 0; t < TPW; ++t)
            acc[t] = __builtin_amdgcn_wmma_f32_16x16x4_f32(
                false, av[t], false, bv, (short)0, acc[t], false, false);
        __syncthreads();
    }

    // ---- emit: [A1(12) | A2(144) | A3(1728) | A4(20736)] per batch ----
    const size_t ob = (size_t)b * 22620;
    if (tid < DD) out[ob + tid] = P[255 * DD + tid] - P[tid];       // A1 = P_last - P_0
    if (tid < 288) {
        if (half == 0) out[ob + DD + i2] = a2own;
#pragma unroll
        for (int rl = 0; rl < 6; ++rl)
            out[ob + 156 + i2 * 12 + rbase + rl] = a3[rl];
    }
    if (n < DD) {
#pragma unroll
        for (int t = 0; t < TPW; ++t) {
            const int m = w * TPW + t;
#pragma unroll
            for (int r = 0; r < 8; ++r) {
                int i3 = 16 * m + r + hi;
                out[ob + 1884 + (size_t)i3 * 12 + n] = acc[t][r];
            }
        }
    }
}

extern "C" void kernel_launch(void* const* d_in, const int* in_sizes, int n_in,
                              void* d_out, int out_size, void* d_ws, size_t ws_size,
                              hipStream_t stream) {
    const float* inp = (const float*)d_in[0];
    float* out = (float*)d_out;
    int nbatch = in_sizes[0] / (LPATH * DD);   // 1024
    hipLaunchKernelGGL(sig_scan_wmma, dim3(nbatch), dim3(TPB), 0, stream,
                       inp, out, nbatch);
}